// ArcDecoder_52578989638036
// MI455X (gfx1250) — compile-verified
//
#include <hip/hip_runtime.h>

typedef float v2f __attribute__((ext_vector_type(2)));
typedef float v4f __attribute__((ext_vector_type(4)));
typedef float v8f __attribute__((ext_vector_type(8)));

#define LSTRIDE 132              // padded row stride (floats): bank = 4*row + col
#define WAVES_PER_BLOCK 4

// Dynamic LDS layout (floats):
//   w1L[128*132] | w2L[128*132] | wbL[128*132] | b1L[128] | b2L[128]
//   per-wave: xs1[16*132] (X1 tile, later reduction scratch)
//             xs2[16*132] (X2 tile, later H2 tile)
#define SMEM_FLOATS (3*128*LSTRIDE + 256 + WAVES_PER_BLOCK*2*16*LSTRIDE)

__global__ __launch_bounds__(128) void arc_decoder_wmma_f32(
    const float* __restrict__ z,  const float* __restrict__ w1, const float* __restrict__ b1,
    const float* __restrict__ w2, const float* __restrict__ b2, const float* __restrict__ wb,
    const float* __restrict__ bb, const int*   __restrict__ arcs,
    float* __restrict__ out, int nEdges)
{
    extern __shared__ float smem[];
    float* w1L = smem;
    float* w2L = smem + 128*LSTRIDE;
    float* wbL = smem + 2*128*LSTRIDE;
    float* b1L = smem + 3*128*LSTRIDE;
    float* b2L = b1L + 128;
    const int wave = threadIdx.x >> 5;
    float* waveBase = b2L + 128 + wave * (2*16*LSTRIDE);
    float* xs1 = waveBase;                 // X1 tile; reused as reduction scratch
    float* xs2 = waveBase + 16*LSTRIDE;    // X2 tile; reused as H2 tile

    // ---- stage weights into LDS (row-major 128x128 -> padded LSTRIDE) ----
    for (int idx = threadIdx.x; idx < 128*32; idx += blockDim.x) {
        const int r = idx >> 5;
        const int c = (idx & 31) << 2;
        *(v4f*)(w1L + r*LSTRIDE + c) = *(const v4f*)(w1 + r*128 + c);
        *(v4f*)(w2L + r*LSTRIDE + c) = *(const v4f*)(w2 + r*128 + c);
        *(v4f*)(wbL + r*LSTRIDE + c) = *(const v4f*)(wb + r*128 + c);
    }
    if (threadIdx.x < 128) {
        b1L[threadIdx.x] = b1[threadIdx.x];
        b2L[threadIdx.x] = b2[threadIdx.x];
    }
    __syncthreads();

    const float bbv    = bb[0];
    const int   lane   = threadIdx.x & 31;
    const int   halfId = lane >> 4;        // 0: lanes 0-15, 1: lanes 16-31
    const int   lr     = lane & 15;
    const int   coff   = halfId << 1;      // k-column offset of this lane's A/B fragment

    const int nTiles     = (nEdges + 15) >> 4;
    const int waveId     = blockIdx.x * WAVES_PER_BLOCK + wave;
    const int waveStride = gridDim.x * WAVES_PER_BLOCK;

    for (int tile = waveId; tile < nTiles; tile += waveStride) {
        const int e0 = tile << 4;

        // ---- gather: lanes 0-15 fetch X1 rows (src), lanes 16-31 fetch X2 rows (dst) ----
        {
            int e = e0 + lr;
            if (e >= nEdges) e = nEdges - 1;             // clamp (tail tiles)
            const int node = arcs[(e << 1) + halfId];    // [:,0]=src, [:,1]=dst
            const float* zr = z + (size_t)(unsigned)node * 128u;
            float* xd = (halfId ? xs2 : xs1) + lr * LSTRIDE;
            #pragma unroll
            for (int c = 0; c < 128; c += 4)
                *(v4f*)(xd + c) = *(const v4f*)(zr + c);
        }

        // ---- prefetch next tile's rows while this tile computes (global_prefetch_b8) ----
        {
            const int nt = tile + waveStride;            // wave-uniform condition
            if (nt < nTiles) {
                int en = (nt << 4) + lr;
                if (en >= nEdges) en = nEdges - 1;
                const int nodeN = arcs[(en << 1) + halfId];
                const float* zp = z + (size_t)(unsigned)nodeN * 128u;
                __builtin_prefetch(zp +  0, 0, 3);       // 512 B row = 4 x 128 B lines
                __builtin_prefetch(zp + 32, 0, 3);
                __builtin_prefetch(zp + 64, 0, 3);
                __builtin_prefetch(zp + 96, 0, 3);
            }
        }
        asm volatile("s_wait_dscnt 0x0" ::: "memory");

        // ---- GEMM2: H2 = relu(X2 @ W2^T + b2), all 8 n-tiles in registers ----
        v8f acc[8];
        #pragma unroll
        for (int t = 0; t < 8; ++t) {
            const float bv = b2L[t*16 + lr];
            acc[t] = {bv, bv, bv, bv, bv, bv, bv, bv};
        }
        #pragma unroll 4
        for (int k = 0; k < 128; k += 4) {
            const v2f a = *(const v2f*)(xs2 + lr*LSTRIDE + k + coff);
            #pragma unroll
            for (int t = 0; t < 8; ++t) {
                const v2f bw = *(const v2f*)(w2L + (t*16 + lr)*LSTRIDE + k + coff);
                acc[t] = __builtin_amdgcn_wmma_f32_16x16x4_f32(
                    false, a, false, bw, (short)0, acc[t], false, false);
            }
        }
        // ReLU + store H2 into xs2 (overwrite X2; LDS ops are in-order per wave)
        #pragma unroll
        for (int t = 0; t < 8; ++t) {
            #pragma unroll
            for (int i = 0; i < 8; ++i) {
                float v = acc[t][i];
                v = v > 0.f ? v : 0.f;
                xs2[(i + (halfId << 3))*LSTRIDE + t*16 + lr] = v;
            }
        }
        asm volatile("s_wait_dscnt 0x0" ::: "memory");

        // ---- per n-tile: H1_t = relu(X1 W1^T + b1), G_t = H2 WB^T; p += H1_t ⊙ G_t ----
        v8f p = {0.f, 0.f, 0.f, 0.f, 0.f, 0.f, 0.f, 0.f};
        #pragma unroll
        for (int t = 0; t < 8; ++t) {
            const float bv = b1L[t*16 + lr];
            v8f c1 = {bv, bv, bv, bv, bv, bv, bv, bv};
            v8f cg = {0.f, 0.f, 0.f, 0.f, 0.f, 0.f, 0.f, 0.f};
            #pragma unroll 4
            for (int k = 0; k < 128; k += 4) {
                const v2f a1  = *(const v2f*)(xs1 + lr*LSTRIDE + k + coff);
                const v2f bw1 = *(const v2f*)(w1L + (t*16 + lr)*LSTRIDE + k + coff);
                c1 = __builtin_amdgcn_wmma_f32_16x16x4_f32(
                    false, a1, false, bw1, (short)0, c1, false, false);
                const v2f a2  = *(const v2f*)(xs2 + lr*LSTRIDE + k + coff);   // H2
                const v2f bwb = *(const v2f*)(wbL + (t*16 + lr)*LSTRIDE + k + coff);
                cg = __builtin_amdgcn_wmma_f32_16x16x4_f32(
                    false, a2, false, bwb, (short)0, cg, false, false);
            }
            #pragma unroll
            for (int i = 0; i < 8; ++i) {
                const float h = c1[i] > 0.f ? c1[i] : 0.f;
                p[i] += h * cg[i];
            }
        }

        // ---- reduce over n within each 16-lane half, via xs1 (X1 no longer needed) ----
        asm volatile("s_wait_dscnt 0x0" ::: "memory");
        #pragma unroll
        for (int i = 0; i < 8; ++i)
            xs1[(halfId << 7) + i*16 + lr] = p[i];      // edge m=i+8*halfId, slice lr
        asm volatile("s_wait_dscnt 0x0" ::: "memory");
        if (lane < 16) {
            const int e = e0 + lane;                     // edge m = lane
            if (e < nEdges) {
                float s = bbv;
                const float* rp = xs1 + ((lane >> 3) << 7) + (lane & 7)*16;
                #pragma unroll
                for (int ss = 0; ss < 16; ++ss) s += rp[ss];
                out[e] = s;
            }
        }
    }
}

extern "C" void kernel_launch(void* const* d_in, const int* in_sizes, int n_in,
                              void* d_out, int out_size, void* d_ws, size_t ws_size,
                              hipStream_t stream) {
    const float* z    = (const float*)d_in[0];
    const float* w1   = (const float*)d_in[1];
    const float* b1   = (const float*)d_in[2];
    const float* w2   = (const float*)d_in[3];
    const float* b2   = (const float*)d_in[4];
    const float* wb   = (const float*)d_in[5];   // (1,128,128)
    const float* bb   = (const float*)d_in[6];   // (1,)
    const int*   arcs = (const int*)d_in[7];     // (E,2) int
    float*       out  = (float*)d_out;

    const int nEdges = out_size;                 // 1,000,000
    const int nTiles = (nEdges + 15) >> 4;
    int blocks = (nTiles + WAVES_PER_BLOCK - 1) / WAVES_PER_BLOCK;
    if (blocks > 2048) blocks = 2048;
    if (blocks < 1)    blocks = 1;

    const size_t shmemBytes = (size_t)SMEM_FLOATS * sizeof(float);  // 271,360 B < 320 KB

    arc_decoder_wmma_f32<<<blocks, 128, shmemBytes, stream>>>(
        z, w1, b1, w2, b2, wb, bb, arcs, out, nEdges);
}